// TemporalNorm_53635551592997
// MI455X (gfx1250) — compile-verified
//
#include <hip/hip_runtime.h>
#include <stdint.h>
#include <math.h>

// ---------------------------------------------------------------------------
// Problem constants (match the JAX reference exactly).
// ---------------------------------------------------------------------------
namespace {
constexpr int kB    = 16;
constexpr int kT    = 4096;
constexpr int kD    = 512;
constexpr int kHalf = 64;              // W/2, window = [t-64, t+63] clipped
constexpr float kEps = 1e-5f;

constexpr int CHUNK  = 256;            // timesteps per thread
constexpr int NCHUNK = kT / CHUNK;     // 16
constexpr int BLOCK  = 256;            // threads per block (8 wave32)
constexpr int DTILES = kD / BLOCK;     // 2

// LDS ring buffer: RING rows of BLOCK floats. Must satisfy RING > 128 + PF.
constexpr int RING  = 160;             // rows resident (128 window + 24 pf + 8 slack)
constexpr int PF    = 24;              // async prefetch depth in rows
constexpr int ROWF  = BLOCK;           // floats per ring row
constexpr int RINGF = RING * ROWF;     // floats in ring
constexpr size_t LDS_BYTES = (size_t)RINGF * 4;   // 160 KB
}  // namespace

// gfx1250 async DMA: global -> LDS, 4B per lane, tracked by ASYNCcnt.
// VDST = LDS byte address VGPR, VADDR = 32-bit byte offset, SADDR = 64-bit base.
#define ASYNC_LOAD_ROW(ldsOff, gOff, base)                                  \
  asm volatile("global_load_async_to_lds_b32 %0, %1, %2"                    \
               :: "v"(ldsOff), "v"(gOff), "s"(base) : "memory")

// ASYNCcnt wait with compile-time immediate. Async loads complete in order,
// so "cnt <= PF" proves the oldest (issued - PF) rows have landed in LDS.
#define WAIT_ASYNC(n) asm volatile("s_wait_asynccnt " #n ::: "memory")

// Sliding-window temporal norm, incoming stream staged into an LDS ring via
// async DMA. Each lane touches only its own column's LDS slots, so per-wave
// ASYNCcnt ordering is sufficient and no workgroup barriers are needed.
__global__ __launch_bounds__(BLOCK)
void TemporalNorm_53635551592997_kernel(const float* __restrict__ x,
                                        const float* __restrict__ weight,
                                        const float* __restrict__ bias,
                                        float* __restrict__ out) {
    extern __shared__ float ring[];

    const int tid   = (int)threadIdx.x;
    const int dtile = blockIdx.x % DTILES;
    const int chunk = (blockIdx.x / DTILES) % NCHUNK;
    const int b     = blockIdx.x / (DTILES * NCHUNK);
    const int d     = dtile * BLOCK + tid;

    const float* __restrict__ xb = x + (size_t)b * kT * kD;        // SGPR base
    float* __restrict__       ob = out + (size_t)b * kT * kD + d;

    const float wgt = weight[d];
    const float bi  = bias[d];

    const int t0 = chunk * CHUNK;
    const int l0 = (t0 - kHalf > 0) ? (t0 - kHalf) : 0;
    const int r0 = (t0 + kHalf - 1 < kT - 1) ? (t0 + kHalf - 1) : (kT - 1);

    // LDS byte address of this lane's column slot in ring row 0.
    const uint32_t ldsBase = (uint32_t)(uintptr_t)ring + (uint32_t)tid * 4u;

    // ---- Warm-up: async-stage rows [l0, min(t0+63+PF, T-1)] into the ring.
    int lastIssue = t0 + kHalf - 1 + PF;
    if (lastIssue > kT - 1) lastIssue = kT - 1;

    uint32_t idxIssue = (uint32_t)(l0 % RING) * ROWF;   // ring row float index
    uint32_t gOff     = ((uint32_t)l0 * kD + (uint32_t)d) * 4u;
    for (int i = l0; i <= lastIssue; ++i) {
        ASYNC_LOAD_ROW(ldsBase + idxIssue * 4u, gOff, xb);
        idxIssue += ROWF; if (idxIssue >= RINGF) idxIssue -= RINGF;
        gOff += (uint32_t)kD * 4u;
    }
    int nextRow = lastIssue + 1;

    // Rows l0..r0 (<=128) are guaranteed complete once outstanding <= PF.
    if (t0 + kHalf - 1 + PF < kT) { WAIT_ASYNC(24); } else { WAIT_ASYNC(0); }

    // ---- Warm-up window sum from LDS.
    float s = 0.0f, s2 = 0.0f;
    {
        uint32_t idx = (uint32_t)(l0 % RING) * ROWF;
        for (int i = l0; i <= r0; ++i) {
            const float v = ring[idx + tid];
            s += v; s2 += v * v;
            idx += ROWF; if (idx >= RINGF) idx -= RINGF;
        }
    }

    // ---- Steady state.
    int l = l0, r = r0;
    uint32_t idxT  = (uint32_t)(t0 % RING) * ROWF;            // row t
    uint32_t idxIn = (uint32_t)((t0 + kHalf) % RING) * ROWF;  // row t+63 @ t=t0+1
    uint32_t idxL  = (uint32_t)(l0 % RING) * ROWF;            // row l

    #pragma unroll 2
    for (int t = t0; t < t0 + CHUNK; ++t) {
        // Issue one more async row (keeps issued-through == t+63+PF).
        if (nextRow < kT) {
            ASYNC_LOAD_ROW(ldsBase + idxIssue * 4u, gOff, xb);
            idxIssue += ROWF; if (idxIssue >= RINGF) idxIssue -= RINGF;
            gOff += (uint32_t)kD * 4u;
            ++nextRow;
        }
        // Ensure rows up to min(t+63, T-1) have landed in LDS.
        if (t + kHalf - 1 + PF < kT) { WAIT_ASYNC(24); } else { WAIT_ASYNC(0); }

        if (t > t0) {
            const int nr = t + kHalf - 1;            // incoming row
            if (nr < kT) {
                const float v = ring[idxIn + tid];
                s += v; s2 += v * v;
                r = nr;
                idxIn += ROWF; if (idxIn >= RINGF) idxIn -= RINGF;
            }
            const int nl = t - kHalf;                // new left edge
            if (nl > l) {
                const float v = ring[idxL + tid];    // element leaving window
                s -= v; s2 -= v * v;
                l = nl;
                idxL += ROWF; if (idxL >= RINGF) idxL -= RINGF;
            }
        }

        const float xt = ring[idxT + tid];
        idxT += ROWF; if (idxT >= RINGF) idxT -= RINGF;

        const float cnt   = (float)(r - l + 1);
        const float mean  = s / cnt;
        const float var   = s2 / cnt - mean * mean;
        const float scale = sqrtf(var + kEps);
        const float y     = (xt - mean) / scale * wgt + bi;

        // Output is never re-read: NT store keeps x resident in the 192MB L2.
        __builtin_nontemporal_store(y, ob + (size_t)t * kD);
    }
}

extern "C" void kernel_launch(void* const* d_in, const int* in_sizes, int n_in,
                              void* d_out, int out_size, void* d_ws, size_t ws_size,
                              hipStream_t stream) {
    (void)in_sizes; (void)n_in; (void)out_size; (void)d_ws; (void)ws_size;

    const float* x      = (const float*)d_in[0];
    const float* weight = (const float*)d_in[1];
    const float* bias   = (const float*)d_in[2];
    float* out          = (float*)d_out;

    const dim3 grid(kB * NCHUNK * DTILES);   // 16 * 16 * 2 = 512 blocks
    const dim3 block(BLOCK);                 // 256 threads = 8 wave32

    TemporalNorm_53635551592997_kernel<<<grid, block, LDS_BYTES, stream>>>(
        x, weight, bias, out);
}